// H2GCNEncoder_38517266710885
// MI455X (gfx1250) — compile-verified
//
#include <hip/hip_runtime.h>
#include <stddef.h>

typedef __attribute__((ext_vector_type(2))) float v2f;
typedef __attribute__((ext_vector_type(8))) float v8f;

#define HID 128
#define KC 32
#define AS_STRIDE 36    // floats; 144B rows: 16B aligned, conflict-free frag reads
#define BS_STRIDE 288   // floats; K-pair-interleaved rows (256 data + 32 pad):
                        // 288 mod 64 = 32 banks -> half-wave b64 bank sets disjoint

// ---------------------------------------------------------------------------
// Utility kernels
// ---------------------------------------------------------------------------
__global__ void zero_f4(float* __restrict__ p, int n4) {
    int i = blockIdx.x * blockDim.x + threadIdx.x;
    if (i < n4) ((float4*)p)[i] = make_float4(0.f, 0.f, 0.f, 0.f);
}

__global__ void deg_kernel(const int* __restrict__ dst, float* __restrict__ deg, int nE) {
    int e = blockIdx.x * blockDim.x + threadIdx.x;
    if (e < nE)
        __hip_atomic_fetch_add(&deg[dst[e]], 1.0f, __ATOMIC_RELAXED, __HIP_MEMORY_SCOPE_AGENT);
}

// one wave per edge; each lane handles 4 consecutive features (32*4 = 128)
__global__ void scatter_add(const float* __restrict__ h, const int* __restrict__ src,
                            const int* __restrict__ dst, float* __restrict__ s, int nE) {
    int t = blockIdx.x * blockDim.x + threadIdx.x;
    int e = t >> 5;
    int lane = t & 31;
    if (e >= nE) return;
    int sr = src[e];
    int dr = dst[e];
    float4 v = *(const float4*)(h + (size_t)sr * HID + lane * 4);
    float* o = s + (size_t)dr * HID + lane * 4;
    __hip_atomic_fetch_add(o + 0, v.x, __ATOMIC_RELAXED, __HIP_MEMORY_SCOPE_AGENT);
    __hip_atomic_fetch_add(o + 1, v.y, __ATOMIC_RELAXED, __HIP_MEMORY_SCOPE_AGENT);
    __hip_atomic_fetch_add(o + 2, v.z, __ATOMIC_RELAXED, __HIP_MEMORY_SCOPE_AGENT);
    __hip_atomic_fetch_add(o + 3, v.w, __ATOMIC_RELAXED, __HIP_MEMORY_SCOPE_AGENT);
}

// s[i][:] /= max(deg[i], 1)
__global__ void normalize_kernel(float* __restrict__ s, const float* __restrict__ deg, int nNodes) {
    int t = blockIdx.x * blockDim.x + threadIdx.x;
    int node = t >> 5;
    int lane = t & 31;
    if (node >= nNodes) return;
    float inv = 1.0f / fmaxf(deg[node], 1.0f);
    float4* p = (float4*)(s + (size_t)node * HID + lane * 4);
    float4 v = *p;
    v.x *= inv; v.y *= inv; v.z *= inv; v.w *= inv;
    *p = v;
}

// ---------------------------------------------------------------------------
// WMMA fp32 GEMM with implicit concat:
//   C[M,128] = [A1 (M x K1) | A2 (M x K2)] @ W[(K1+K2) x 128] + bias, optional relu
// Block: 256 threads = 8 waves; block tile 128 rows x 128 cols; wave = 16-row strip.
// B staged K-pair interleaved so each B fragment is one contiguous ds_load_b64.
// ---------------------------------------------------------------------------
__global__ __launch_bounds__(256) void gemm_wmma_cat(
    const float* __restrict__ A1, int K1,
    const float* __restrict__ A2, int K2,
    const float* __restrict__ W,
    const float* __restrict__ bias,
    float* __restrict__ C,
    int M, int relu)
{
    __shared__ float As[128][AS_STRIDE];
    __shared__ float Bs[KC / 2][BS_STRIDE];   // Bs[k>>1][n*2 + (k&1)]

    const int tid  = threadIdx.x;
    const int lane = tid & 31;
    const int wid  = tid >> 5;       // 0..7
    const int half = lane >> 4;      // 0/1: K-high half of the wave
    const int l16  = lane & 15;
    const int m0   = blockIdx.x * 128;
    const int K    = K1 + K2;

    v8f acc[8];
#pragma unroll
    for (int i = 0; i < 8; ++i)
#pragma unroll
        for (int j = 0; j < 8; ++j) acc[i][j] = 0.0f;

    for (int k0 = 0; k0 < K; k0 += KC) {
        const float* Ap;
        int lda, kc;
        if (k0 < K1) { Ap = A1; lda = K1; kc = k0; }
        else         { Ap = A2; lda = K2; kc = k0 - K1; }

        // stage A chunk: 128 rows x 32 cols = 1024 float4 (4 per thread)
#pragma unroll
        for (int i = 0; i < 4; ++i) {
            int e  = tid + i * 256;  // 0..1023
            int r  = e >> 3;         // 0..127
            int c4 = e & 7;          // 0..7
            float4 v = make_float4(0.f, 0.f, 0.f, 0.f);
            if (m0 + r < M)
                v = *(const float4*)(Ap + (size_t)(m0 + r) * lda + kc + c4 * 4);
            *(float4*)(&As[r][c4 * 4]) = v;
        }
        // stage B chunk K-pair interleaved: 16 pair-rows x 64 col-pairs = 1024 float4
        // Bs[pr][2c..2c+3] = { W[2pr][c], W[2pr+1][c], W[2pr][c+1], W[2pr+1][c+1] }
#pragma unroll
        for (int i = 0; i < 4; ++i) {
            int e  = tid + i * 256;  // 0..1023
            int pr = e >> 6;         // 0..15 pair-row
            int c2 = e & 63;         // 0..63 column pair
            float2 g0 = *(const float2*)(W + (size_t)(k0 + 2 * pr + 0) * HID + c2 * 2);
            float2 g1 = *(const float2*)(W + (size_t)(k0 + 2 * pr + 1) * HID + c2 * 2);
            *(float4*)(&Bs[pr][c2 * 4]) = make_float4(g0.x, g1.x, g0.y, g1.y);
        }
        __syncthreads();

#pragma unroll
        for (int kk = 0; kk < KC; kk += 4) {
            // A frag 16x4: lane l holds A[m0+wid*16 + l16][kk + half*2 + v]
            v2f a = *(const v2f*)(&As[wid * 16 + l16][kk + half * 2]);
            const int pr = (kk >> 1) + half;  // pair-row holding K = kk+half*2, +1
#pragma unroll
            for (int ct = 0; ct < 8; ++ct) {
                // B frag 4x16: contiguous K-pair -> single b64 LDS read
                v2f b = *(const v2f*)(&Bs[pr][(ct * 16 + l16) * 2]);
                acc[ct] = __builtin_amdgcn_wmma_f32_16x16x4_f32(
                    false, a, false, b, (short)0, acc[ct], false, false);
            }
        }
        __syncthreads();
    }

    const int mbase = m0 + wid * 16;
    if (mbase < M) {  // M % 16 == 0 -> whole strip valid
#pragma unroll
        for (int ct = 0; ct < 8; ++ct) {
            int col = ct * 16 + l16;
            float bv = bias[col];
#pragma unroll
            for (int r = 0; r < 8; ++r) {
                int row = mbase + half * 8 + r;  // C/D layout: vgpr r -> M = r + half*8
                float v = acc[ct][r] + bv;
                if (relu) v = fmaxf(v, 0.0f);
                C[(size_t)row * HID + col] = v;
            }
        }
    }
}

// ---------------------------------------------------------------------------
// Host-side orchestration
// ---------------------------------------------------------------------------
static inline size_t align256(size_t x) { return (x + 255) & ~(size_t)255; }

extern "C" void kernel_launch(void* const* d_in, const int* in_sizes, int n_in,
                              void* d_out, int out_size, void* d_ws, size_t ws_size,
                              hipStream_t stream) {
    const float* x     = (const float*)d_in[0];
    const float* W_in  = (const float*)d_in[1];
    const float* b_in  = (const float*)d_in[2];
    const float* W_m0  = (const float*)d_in[3];
    const float* b_m0  = (const float*)d_in[4];
    const float* W_m1  = (const float*)d_in[5];
    const float* b_m1  = (const float*)d_in[6];
    const float* W_out = (const float*)d_in[7];
    const float* b_out = (const float*)d_in[8];
    const int*   ei    = (const int*)d_in[9];

    const int IN_DIM = 256;
    const int N  = in_sizes[0] / IN_DIM;   // 40000 nodes
    const int nE = in_sizes[9] / 2;        // 640000 edges
    const int* src = ei;
    const int* dst = ei + nE;

    char* ws = (char*)d_ws;
    size_t off = 0;
    float* deg = (float*)(ws + off); off += align256((size_t)N * sizeof(float));
    float* h   = (float*)(ws + off); off += align256((size_t)N * HID * sizeof(float));
    float* h1  = (float*)(ws + off); off += align256((size_t)N * HID * sizeof(float));
    float* h2  = (float*)(ws + off); off += align256((size_t)N * HID * sizeof(float));
    (void)ws_size; (void)n_in; (void)out_size;

    const int B = 256;
    const int degZ  = (N / 4 + B - 1) / B;
    const int degB  = (nE + B - 1) / B;
    const int sumZ  = (N * HID / 4 + B - 1) / B;
    const int scatB = (int)(((size_t)nE * 32 + B - 1) / B);
    const int normB = (int)(((size_t)N * 32 + B - 1) / B);
    const int gemmB = (N + 127) / 128;

    // degrees (once; reused by all 4 aggregations)
    zero_f4<<<degZ, B, 0, stream>>>(deg, N / 4);
    deg_kernel<<<degB, B, 0, stream>>>(dst, deg, nE);

    // h = relu(x @ W_in + b_in)
    gemm_wmma_cat<<<gemmB, B, 0, stream>>>(x, IN_DIM, nullptr, 0, W_in, b_in, h, N, 1);

    const float* Wmix[2] = {W_m0, W_m1};
    const float* bmix[2] = {b_m0, b_m1};
    for (int layer = 0; layer < 2; ++layer) {
        // h1 = mean_agg(h)
        zero_f4<<<sumZ, B, 0, stream>>>(h1, N * HID / 4);
        scatter_add<<<scatB, B, 0, stream>>>(h, src, dst, h1, nE);
        normalize_kernel<<<normB, B, 0, stream>>>(h1, deg, N);
        // h2 = mean_agg(h1)
        zero_f4<<<sumZ, B, 0, stream>>>(h2, N * HID / 4);
        scatter_add<<<scatB, B, 0, stream>>>(h1, src, dst, h2, nE);
        normalize_kernel<<<normB, B, 0, stream>>>(h2, deg, N);
        // h = [h1|h2] @ W_mix + b_mix  (relu on layer 0 only)
        gemm_wmma_cat<<<gemmB, B, 0, stream>>>(h1, HID, h2, HID, Wmix[layer], bmix[layer],
                                               h, N, layer == 0 ? 1 : 0);
    }

    // out = h @ W_out + b_out
    gemm_wmma_cat<<<gemmB, B, 0, stream>>>(h, HID, nullptr, 0, W_out, b_out,
                                           (float*)d_out, N, 0);
}